// deform_conv_2d_74156905332894
// MI455X (gfx1250) — compile-verified
//
#include <hip/hip_runtime.h>
#include <hip/hip_bf16.h>

typedef __attribute__((ext_vector_type(2))) float v2f;
typedef __attribute__((ext_vector_type(8))) float v8f;

#define BATCH 4
#define CIN   64
#define COUT  64
#define HH    128
#define WW    128
#define K2    9
#define HW    (HH * WW)

// ---------------------------------------------------------------------------
// Kernel 1: offset conv  x(B,64,128,128) * w_offset(18,64,3,3) -> off(B,18,128,128)
// grid (Ho, B), block 128 (one thread per wo). Weights staged in LDS (41.5 KB).
// ---------------------------------------------------------------------------
__global__ __launch_bounds__(128) void dc_offset_conv(
    const float* __restrict__ x, const float* __restrict__ w_off,
    float* __restrict__ off)
{
    __shared__ float ws[18 * 64 * 9];
    const int tid = threadIdx.x;
    for (int i = tid; i < 18 * 64 * 9; i += 128) ws[i] = w_off[i];
    __syncthreads();

    const int ho = blockIdx.x;
    const int b  = blockIdx.y;
    const int wo = tid;

    float acc[18];
#pragma unroll
    for (int ch = 0; ch < 18; ++ch) acc[ch] = 0.f;

    const float* xb = x + (size_t)b * CIN * HW;
    for (int c = 0; c < CIN; ++c) {
        const float* xc = xb + c * HW;
#pragma unroll
        for (int ky = 0; ky < 3; ++ky) {
            const int y = ho - 1 + ky;
            if (y < 0 || y >= HH) continue;
            float xv[3];
#pragma unroll
            for (int kx = 0; kx < 3; ++kx) {
                const int xx = wo - 1 + kx;
                xv[kx] = (xx >= 0 && xx < WW) ? xc[y * WW + xx] : 0.f;
            }
#pragma unroll
            for (int ch = 0; ch < 18; ++ch) {
                const float* wp = &ws[((ch * 64 + c) * 3 + ky) * 3];
                acc[ch] += xv[0] * wp[0] + xv[1] * wp[1] + xv[2] * wp[2];
            }
        }
    }

    float* ob = off + (((size_t)b * 18) * HH + ho) * WW + wo;
#pragma unroll
    for (int ch = 0; ch < 18; ++ch) ob[(size_t)ch * HW] = acc[ch];
}

// ---------------------------------------------------------------------------
// Kernel 2: fused bilinear sampling + WMMA GEMM.
// out[b,o,h,w] = sum_{c,k2} v[b,c,k2,h,w] * wd[o,c,k2]
// One workgroup per (ho, b): 128 pixels. M=cout(64), N=pixel(128), K=576.
// K split as 9 slices of 64 channels; each slice staged in LDS then consumed
// by fp32 WMMA (v_wmma_f32_16x16x4_f32). 8 waves: wave w owns N-tile w,
// accumulating all 4 M-tiles.
// ---------------------------------------------------------------------------
__global__ __launch_bounds__(256) void dc_sample_gemm(
    const float* __restrict__ x, const float* __restrict__ off,
    const float* __restrict__ wd, float* __restrict__ out)
{
    __shared__ float Ws[64][68];    // weight slice [cout][c], padded (stride 68)
    __shared__ float Vs[64][132];   // sampled tile [c][pixel], padded (stride 132)

    const int tid  = threadIdx.x;
    const int lane = tid & 31;
    const int wv   = tid >> 5;          // wave id 0..7
    const int half = lane >> 4;         // 0: lanes 0-15, 1: lanes 16-31
    const int l    = lane & 15;

    const int ho = blockIdx.x;
    const int b  = blockIdx.y;

    const int p  = tid & 127;           // pixel handled by this thread (sampling)
    const int cb = (tid >> 7) * 32;     // channel base (0 or 32)

    const float* xb = x + (size_t)b * CIN * HW;

    v8f acc[4];
#pragma unroll
    for (int m = 0; m < 4; ++m)
#pragma unroll
        for (int r = 0; r < 8; ++r) acc[m][r] = 0.f;

    for (int k2 = 0; k2 < K2; ++k2) {
        __syncthreads();   // previous slice fully consumed before overwrite

        // --- stage weight slice: Ws[o][c] = wd[(o*64+c)*9 + k2] ---
        for (int i = tid; i < 64 * 64; i += 256) {
            const int o = i >> 6, c = i & 63;
            Ws[o][c] = wd[(size_t)(o * 64 + c) * 9 + k2];
        }

        // --- bilinear sampling params for this pixel ---
        const int ky = k2 / 3, kx = k2 % 3;
        const float dy = off[(((size_t)b * 18 + 2 * k2    ) * HH + ho) * WW + p];
        const float dx = off[(((size_t)b * 18 + 2 * k2 + 1) * HH + ho) * WW + p];
        const float py = (float)(ho - 1 + ky) + dy;
        const float px = (float)(p  - 1 + kx) + dx;
        const float y0f = floorf(py), x0f = floorf(px);
        const float ly = py - y0f,    lx = px - x0f;
        const int y0 = (int)y0f, x0 = (int)x0f;
        const int y1 = y0 + 1,   x1 = x0 + 1;

        const float vy0 = (y0 >= 0 && y0 < HH) ? 1.f : 0.f;
        const float vy1 = (y1 >= 0 && y1 < HH) ? 1.f : 0.f;
        const float vx0 = (x0 >= 0 && x0 < WW) ? 1.f : 0.f;
        const float vx1 = (x1 >= 0 && x1 < WW) ? 1.f : 0.f;

        const float w00 = (1.f - ly) * (1.f - lx) * vy0 * vx0;
        const float w01 = (1.f - ly) * lx         * vy0 * vx1;
        const float w10 = ly * (1.f - lx)         * vy1 * vx0;
        const float w11 = ly * lx                 * vy1 * vx1;

        const int y0c = min(max(y0, 0), HH - 1), y1c = min(max(y1, 0), HH - 1);
        const int x0c = min(max(x0, 0), WW - 1), x1c = min(max(x1, 0), WW - 1);
        const int a00 = y0c * WW + x0c, a01 = y0c * WW + x1c;
        const int a10 = y1c * WW + x0c, a11 = y1c * WW + x1c;

        // --- sample 32 channels for this pixel into Vs[c][p] ---
#pragma unroll 4
        for (int c = 0; c < 32; ++c) {
            const int cc = cb + c;
            const float* xc = xb + (size_t)cc * HW;
            Vs[cc][p] = w00 * xc[a00] + w01 * xc[a01]
                      + w10 * xc[a10] + w11 * xc[a11];
        }

        __syncthreads();   // slice staged; run WMMA on it

        // --- fp32 WMMA over this 64-deep K slice ---
        const int nb = wv * 16;
#pragma unroll
        for (int kk = 0; kk < 64; kk += 4) {
            const int kl = kk + half * 2;   // lanes 0-15: K=kk,kk+1; 16-31: kk+2,kk+3
            v2f bf;                         // B 4x16: v[k][pixel]
            bf.x = Vs[kl][nb + l];
            bf.y = Vs[kl + 1][nb + l];
#pragma unroll
            for (int m = 0; m < 4; ++m) {
                v2f af;                     // A 16x4: W[cout][k]
                af.x = Ws[m * 16 + l][kl];
                af.y = Ws[m * 16 + l][kl + 1];
                acc[m] = __builtin_amdgcn_wmma_f32_16x16x4_f32(
                    /*neg_a=*/false, af, /*neg_b=*/false, bf,
                    /*c_mod=*/(short)0, acc[m],
                    /*reuse_a=*/false, /*reuse_b=*/false);
            }
        }
    }

    // --- store: D vgpr r -> M = m*16 + r (+8 for hi half-lanes), N = pixel ---
    float* ob = out + (((size_t)b * COUT) * HH + ho) * WW;
    const int pix = wv * 16 + l;
#pragma unroll
    for (int m = 0; m < 4; ++m) {
#pragma unroll
        for (int r = 0; r < 8; ++r) {
            const int o = m * 16 + half * 8 + r;
            ob[(size_t)o * HW + pix] = acc[m][r];
        }
    }
}

// ---------------------------------------------------------------------------
extern "C" void kernel_launch(void* const* d_in, const int* in_sizes, int n_in,
                              void* d_out, int out_size, void* d_ws, size_t ws_size,
                              hipStream_t stream) {
    const float* x        = (const float*)d_in[0];   // (4,64,128,128)
    const float* w_offset = (const float*)d_in[1];   // (18,64,3,3)
    const float* w_deform = (const float*)d_in[2];   // (64,64,3,3)
    float*       out      = (float*)d_out;           // (4,64,128,128)
    float*       offbuf   = (float*)d_ws;            // (4,18,128,128) = 4.72 MB

    dim3 grid(HH, BATCH);
    dc_offset_conv<<<grid, 128, 0, stream>>>(x, w_offset, offbuf);
    dc_sample_gemm<<<grid, 256, 0, stream>>>(x, offbuf, w_deform, out);
}